// BertMultiPooler_30434138260161
// MI455X (gfx1250) — compile-verified
//
#include <hip/hip_runtime.h>
#include <hip/hip_bf16.h>
#include <math.h>

// ---------------------------------------------------------------------------
// BertMultiPooler for MI455X (gfx1250, wave32, WMMA + TDM)
// Stage 1: segment mean-pool + cls gather  -> X = [pooled | tab]  (2048 x 1536)
// Stage 2: transpose [W_dense; W_tab]      -> Wc (1536 x 768, K-major)
// Stage 3: fp32 WMMA GEMM (M=2048,N=768,K=1536), TDM double-buffered LDS,
//          fused bias + tanh epilogue
// ---------------------------------------------------------------------------

typedef __attribute__((ext_vector_type(2))) float        v2f;
typedef __attribute__((ext_vector_type(8))) float        v8f;
typedef __attribute__((ext_vector_type(4))) unsigned int v4u;
typedef __attribute__((ext_vector_type(4))) int          v4i;
typedef __attribute__((ext_vector_type(8))) int          v8i;

constexpr int Bn = 32;            // batches
constexpr int Sn = 4096;          // sequence length
constexpr int Hn = 768;           // hidden
constexpr int Kn = 64;            // segments per batch
constexpr int Mt = Bn * Kn;       // 2048 rows
constexpr int KD = 2 * Hn;        // 1536 reduction dim (pooled | tab)
constexpr int Nt = Hn;            // 768 cols

// GEMM tiling
constexpr int BM = 64, BN = 64, BK = 32;
constexpr int ASTR = 49;  // sA row stride: 49^-1=17 mod 64 -> lo/hi half-wave banks disjoint
constexpr int BSTR = 72;  // sB row stride: 2*72 mod 64 = 16 -> lo/hi half-wave banks disjoint

#if defined(__has_builtin)
#if __has_builtin(__builtin_amdgcn_tensor_load_to_lds) && \
    __has_builtin(__builtin_amdgcn_s_wait_tensorcnt)
#define HAVE_TDM 1
#endif
#endif

// ---------------------------------------------------------------------------
// Stage 1: one block per (b,k) segment. Mean over tokens [pos[k], end) and
// gather of the cls row; writes the 1536-wide concatenated activation row.
// ---------------------------------------------------------------------------
__launch_bounds__(256)
__global__ void pool_gather_kernel(const float* __restrict__ hs,
                                   const long long* __restrict__ cls,   // (M,2) i64
                                   const long long* __restrict__ tlen,  // (B,)  i64
                                   float* __restrict__ X) {
    const int m   = blockIdx.x;        // b*Kn + k
    const int b   = m >> 6;            // / Kn
    const int k   = m & (Kn - 1);
    const int tid = threadIdx.x;

    const long long start = cls[2 * m + 1];
    long long end = (k == Kn - 1) ? (long long)Sn : cls[2 * (m + 1) + 1];
    const long long len = tlen[b];
    if (end > len) end = len;
    long long cnt = end - start;
    if (cnt < 0) cnt = 0;
    const float inv = 1.0f / (float)cnt;   // cnt==0 -> 0*inf = NaN, matches ref 0/0

    const float* base = hs + (size_t)b * Sn * Hn;
    float a0 = 0.f, a1 = 0.f, a2 = 0.f;
    for (long long t = start; t < end; ++t) {
        const float* row = base + (size_t)t * Hn;
        __builtin_prefetch(row + Hn, 0, 0);          // global_prefetch_b8
        a0 += row[tid];
        a1 += row[tid + 256];
        a2 += row[tid + 512];
    }

    float* xr = X + (size_t)m * KD;
    xr[tid]       = a0 * inv;
    xr[tid + 256] = a1 * inv;
    xr[tid + 512] = a2 * inv;

    const float* tr = base + (size_t)start * Hn;     // cls gather row
    xr[Hn + tid]       = tr[tid];
    xr[Hn + tid + 256] = tr[tid + 256];
    xr[Hn + tid + 512] = tr[tid + 512];
}

// ---------------------------------------------------------------------------
// Stage 2: Wc[h][n] = (h < H) ? W_dense[n][h] : W_tab[n][h-H]  (K-major weights)
// ---------------------------------------------------------------------------
__launch_bounds__(256)
__global__ void weight_transpose_kernel(const float* __restrict__ Wd,
                                        const float* __restrict__ Wt,
                                        float* __restrict__ Wc) {
    __shared__ float tile[32][33];
    const int th = blockIdx.x;   // h-tile (48)
    const int tn = blockIdx.y;   // n-tile (24)
    const int lx = threadIdx.x & 31;
    const int ly = threadIdx.x >> 5;

    #pragma unroll
    for (int i = 0; i < 32; i += 8) {
        const int n = tn * 32 + ly + i;
        const int h = th * 32 + lx;
        tile[ly + i][lx] = (h < Hn) ? Wd[(size_t)n * Hn + h]
                                    : Wt[(size_t)n * Hn + (h - Hn)];
    }
    __syncthreads();
    #pragma unroll
    for (int i = 0; i < 32; i += 8) {
        const int h = th * 32 + ly + i;
        const int n = tn * 32 + lx;
        Wc[(size_t)h * Nt + n] = tile[lx][ly + i];
    }
}

// ---------------------------------------------------------------------------
// TDM: issue one 2-D tile load Global->LDS via Tensor Data Mover.
// Descriptor built per CDNA5 ISA ch.8 (D# groups 0..3).  pad_interval /
// pad_amount reproduce the bank-conflict-free LDS row strides.
// ---------------------------------------------------------------------------
#ifdef HAVE_TDM
__device__ __forceinline__ void tdm_load_2d(unsigned lds_addr, const void* gptr,
                                            unsigned tensor_d0, unsigned tensor_d1,
                                            unsigned tile_d0, unsigned tile_d1,
                                            unsigned stride0,
                                            unsigned pad_interval,   // pads every 2^(v+1) dwords
                                            unsigned pad_amount) {   // pad of (v+1) dwords
    const unsigned long long ga = (unsigned long long)gptr;
    v4u g0;
    g0.x = 1u;                                         // count=1, user-mode, no gather
    g0.y = lds_addr;                                   // LDS byte address (low 32b of flat)
    g0.z = (unsigned)(ga & 0xFFFFFFFFu);               // global_addr[31:0]
    g0.w = (unsigned)((ga >> 32) & 0x01FFFFFFu)        // global_addr[56:32]
           | (2u << 30);                               // type = 2 ("image")
    v8i g1;
    g1[0] = (int)((2u << 16)                           // data_size = 4 bytes
                  | (1u << 20)                         // pad_enable
                  | (pad_interval << 22)
                  | (pad_amount << 25));
    g1[1] = (int)((tensor_d0 & 0xFFFFu) << 16);        // tensor_dim0[15:0]
    g1[2] = (int)((tensor_d0 >> 16) | ((tensor_d1 & 0xFFFFu) << 16));
    g1[3] = (int)((tensor_d1 >> 16) | (tile_d0 << 16));
    g1[4] = (int)tile_d1;                              // tile_dim2 = 0 (2-D)
    g1[5] = (int)stride0;                              // tensor_dim0_stride[31:0]
    g1[6] = 0;                                         // stride hi / dim1_stride (unused, 2-D)
    g1[7] = 0;
    const v4i z4 = {0, 0, 0, 0};                       // groups 2/3: 2-D tensor
#if __clang_major__ >= 23
    const v8i z8 = {0, 0, 0, 0, 0, 0, 0, 0};
    __builtin_amdgcn_tensor_load_to_lds(g0, g1, z4, z4, z8, 0);
#else
    __builtin_amdgcn_tensor_load_to_lds(g0, g1, z4, z4, 0);
#endif
}
#endif

// ---------------------------------------------------------------------------
// Stage 3: out = tanh(X @ Wc + b_dense + b_tab), fp32 WMMA 16x16x4.
// 64x64 block tile, 8 waves (4 x M, 2 x N), TDM double-buffered K chunks.
// ---------------------------------------------------------------------------
__launch_bounds__(256)
__global__ void gemm_tanh_kernel(const float* __restrict__ X,
                                 const float* __restrict__ Wc,
                                 const float* __restrict__ bd,
                                 const float* __restrict__ bt,
                                 float* __restrict__ out) {
    __shared__ float sA[2][BM][ASTR];   // 2 x 12.25 KB
    __shared__ float sB[2][BK][BSTR];   // 2 x  9.00 KB

    const int tid  = threadIdx.x;
    const int bm0  = blockIdx.x * BM;
    const int bn0  = blockIdx.y * BN;
    const int wave = tid >> 5;
    const int lane = tid & 31;
    const int hi   = lane >> 4;         // half-wave select (K split)
    const int ln   = lane & 15;
    const int wm   = (wave & 3) * 16;   // wave row tile
    const int wn   = (wave >> 2) * 32;  // wave col base (two 16-wide tiles)

    constexpr int NCH = KD / BK;        // 48 K-chunks

    v8f c0 = {};
    v8f c1 = {};

#ifdef HAVE_TDM
    const unsigned ldsA[2] = {(unsigned)(uintptr_t)&sA[0][0][0],
                              (unsigned)(uintptr_t)&sA[1][0][0]};
    const unsigned ldsB[2] = {(unsigned)(uintptr_t)&sB[0][0][0],
                              (unsigned)(uintptr_t)&sB[1][0][0]};

    // Prologue: wave 0 kicks off chunk 0 into buffer 0.
    if (wave == 0) {
        // A tile: rows bm0..bm0+63 of X, cols 0..31.  32-dword rows, stride->49.
        tdm_load_2d(ldsA[0], X + (size_t)bm0 * KD, KD, Mt, BK, BM, KD, 4u, 16u);
        // B tile: rows 0..31 of Wc, cols bn0..bn0+63. 64-dword rows, stride->72.
        tdm_load_2d(ldsB[0], Wc + bn0, Nt, KD, BN, BK, Nt, 5u, 7u);
    }
#endif

    for (int ch = 0; ch < NCH; ++ch) {
        const int cur = ch & 1;
        const int kc  = ch * BK;

#ifdef HAVE_TDM
        if (wave == 0) {
            if (ch + 1 < NCH) {
                const int nkc = kc + BK;
                tdm_load_2d(ldsA[cur ^ 1], X + (size_t)bm0 * KD + nkc,
                            KD, Mt, BK, BM, KD, 4u, 16u);
                tdm_load_2d(ldsB[cur ^ 1], Wc + (size_t)nkc * Nt + bn0,
                            Nt, KD, BN, BK, Nt, 5u, 7u);
                __builtin_amdgcn_s_wait_tensorcnt(2);   // chunk `ch` finished (in-order)
            } else {
                __builtin_amdgcn_s_wait_tensorcnt(0);
            }
        }
        __syncthreads();                                // chunk `ch` visible to all waves
#else
        // Fallback: synchronous cooperative fill of buffer `cur`.
        #pragma unroll
        for (int i = 0; i < (BM * BK) / 256; ++i) {
            const int e = tid + i * 256;
            sA[cur][e >> 5][e & 31] = X[(size_t)(bm0 + (e >> 5)) * KD + kc + (e & 31)];
        }
        #pragma unroll
        for (int i = 0; i < (BK * BN) / 256; ++i) {
            const int e = tid + i * 256;
            sB[cur][e >> 6][e & 63] = Wc[(size_t)(kc + (e >> 6)) * Nt + bn0 + (e & 63)];
        }
        __syncthreads();
#endif

        #pragma unroll
        for (int ks = 0; ks < BK; ks += 4) {
            // A frag: a[v] = A[m][ks + v + 2*hi]   (16x4 fp32 layout)
            v2f a, b0, b1;
            a.x  = sA[cur][wm + ln][ks + 2 * hi];
            a.y  = sA[cur][wm + ln][ks + 2 * hi + 1];
            // B frag: b[v] = B[ks + v + 2*hi][n]   (4x16 fp32 layout)
            b0.x = sB[cur][ks + 2 * hi][wn + ln];
            b0.y = sB[cur][ks + 2 * hi + 1][wn + ln];
            b1.x = sB[cur][ks + 2 * hi][wn + 16 + ln];
            b1.y = sB[cur][ks + 2 * hi + 1][wn + 16 + ln];

            c0 = __builtin_amdgcn_wmma_f32_16x16x4_f32(false, a, false, b0,
                                                       (short)0, c0, false, false);
            c1 = __builtin_amdgcn_wmma_f32_16x16x4_f32(false, a, false, b1,
                                                       (short)0, c1, false, false);
        }
        __syncthreads();   // buffer `cur` free before it is re-targeted at ch+2
    }

    // Epilogue: row = vgpr + 8*hi, col = ln (16x16 f32 C/D layout); bias + tanh
    #pragma unroll
    for (int i = 0; i < 8; ++i) {
        const int gm  = bm0 + wm + i + 8 * hi;
        const int gn0 = bn0 + wn + ln;
        const int gn1 = gn0 + 16;
        out[(size_t)gm * Nt + gn0] = tanhf(c0[i] + bd[gn0] + bt[gn0]);
        out[(size_t)gm * Nt + gn1] = tanhf(c1[i] + bd[gn1] + bt[gn1]);
    }
}

// ---------------------------------------------------------------------------
extern "C" void kernel_launch(void* const* d_in, const int* in_sizes, int n_in,
                              void* d_out, int out_size, void* d_ws, size_t ws_size,
                              hipStream_t stream) {
    const float*     hs   = (const float*)d_in[0];      // (B,S,H) f32
    const float*     Wd   = (const float*)d_in[1];      // (H,H)   f32
    const float*     bd   = (const float*)d_in[2];      // (H,)    f32
    const float*     Wt   = (const float*)d_in[3];      // (H,H)   f32
    const float*     bt   = (const float*)d_in[4];      // (H,)    f32
    const long long* cls  = (const long long*)d_in[5];  // (B*K,2) i64
    const long long* tlen = (const long long*)d_in[6];  // (B,)    i64
    float*           outp = (float*)d_out;              // (B*K,H) f32

    float* X  = (float*)d_ws;                           // 2048*1536 f32 = 12.6 MB
    float* Wc = X + (size_t)Mt * KD;                    // 1536*768  f32 =  4.7 MB

    pool_gather_kernel<<<dim3(Mt), dim3(256), 0, stream>>>(hs, cls, tlen, X);
    weight_transpose_kernel<<<dim3(KD / 32, Nt / 32), dim3(256), 0, stream>>>(Wd, Wt, Wc);
    gemm_tanh_kernel<<<dim3(Mt / BM, Nt / BN), dim3(256), 0, stream>>>(X, Wc, bd, bt, outp);
}